// Grouping_38826504356333
// MI455X (gfx1250) — compile-verified
//
#include <hip/hip_runtime.h>

// Segment-mean (contiguous ragged groups) for B=8, S=8192, H=256, G=1024.
//
// Pass 1: zero out + counts.
// Pass 2: global per-group counts from segment_ids (wave-aggregated runs).
// Pass 3: per-wave 16-row window; scaled one-hot matrix P (P[m,row] =
//         1/count(group_m) if rank(row)==m) times feature window F (16xH)
//         via chained V_WMMA_F32_16X16X4_F32 -> atomics accumulate the MEAN
//         directly into out[b,g,h]. No finalize pass needed.

typedef float v2f __attribute__((ext_vector_type(2)));
typedef float v8f __attribute__((ext_vector_type(8)));

constexpr int Bdim = 8;
constexpr int Sdim = 8192;
constexpr int Hdim = 256;
constexpr int Gdim = 1024;
constexpr int ROWS_PER_WIN = 16;
constexpr int WIN_PER_BATCH = Sdim / ROWS_PER_WIN;          // 512
constexpr int WINDOWS = Bdim * WIN_PER_BATCH;               // 4096
constexpr int WAVES_PER_BLOCK = 8;                          // 256 threads

__global__ void grouping_zero_kernel(float* __restrict__ out,
                                     float* __restrict__ counts,
                                     int nOut, int nCnt) {
    int i = blockIdx.x * blockDim.x + threadIdx.x;
    int stride = gridDim.x * blockDim.x;
    for (int k = i; k < nOut; k += stride) out[k] = 0.0f;
    for (int k = i; k < nCnt; k += stride) counts[k] = 0.0f;
}

// One wave handles 32 consecutive positions of one batch row (S % 32 == 0).
// Run leaders (found via ballot) add their run length: ~1 atomic per run
// instead of 1 per element on these sorted, heavily-colliding ids.
__global__ __launch_bounds__(256)
void grouping_counts_kernel(const int* __restrict__ seg,
                            float* __restrict__ counts) {
    const int lane  = threadIdx.x & 31;
    const int chunk = (blockIdx.x * blockDim.x + threadIdx.x) >> 5;
    const int base  = chunk * 32;                 // within one batch row
    const int b     = base / Sdim;

    const int sv   = seg[base + lane];
    const int prev = __shfl(sv, (lane == 0) ? 0 : (lane - 1), 32);
    const int lead = (lane == 0) || (sv != prev);
    unsigned mask = (unsigned)(__ballot(lead != 0) & 0xFFFFFFFFull);
    if (lead) {
        unsigned above = mask & ~((2u << lane) - 1u);  // leaders after me
        int next = above ? (__ffs(above) - 1) : 32;
        atomicAdd(&counts[b * Gdim + sv], (float)(next - lane));
    }
}

__global__ __launch_bounds__(256)
void grouping_segmean_wmma_kernel(const float* __restrict__ feats,
                                  const int*   __restrict__ seg,
                                  float* __restrict__ out,
                                  const float* __restrict__ counts) {
    const int lane = threadIdx.x & 31;            // wave32
    const int wave = threadIdx.x >> 5;
    const int win  = blockIdx.x * WAVES_PER_BLOCK + wave;
    if (win >= WINDOWS) return;                   // wave-uniform

    const int b       = win / WIN_PER_BATCH;
    const int rowBase = (win % WIN_PER_BATCH) * ROWS_PER_WIN;

    const int r15 = lane & 15;                    // window-row this lane represents
    const int kh  = lane >> 4;                    // wave half (selects K pairs)

    // --- distinct-group bookkeeping (sorted ids => contiguous runs) ---------
    const int sv     = seg[b * Sdim + rowBase + r15];
    const int svPrev = __shfl(sv, (r15 == 0) ? 0 : (r15 - 1), 32);
    const int isNew  = (r15 == 0) ? 1 : ((sv != svPrev) ? 1 : 0);
    unsigned long long bl = __ballot(isNew != 0); // halves are duplicates
    const unsigned mask16 = (unsigned)(bl & 0xFFFFull);
    const int nm   = __popc(mask16);                           // #distinct groups
    const int rank = __popc(mask16 & ((2u << r15) - 1u)) - 1;  // rank of row r15

    // lane m (m = r15) finds the first row of the m-th distinct group
    int firstRow = 0;
    {
        unsigned tmp = mask16;
        for (int k = 0; k <= r15; ++k) {
            if (tmp) { firstRow = __ffs(tmp) - 1; tmp &= tmp - 1; }
        }
    }
    const int myGroup = __shfl(sv, firstRow, 32); // valid when r15 < nm

    // reciprocal global count for my distinct group -> scaled one-hot A
    float inv = 0.0f;
    if (r15 < nm) inv = 1.0f / counts[b * Gdim + myGroup];

    // --- build scaled one-hot A (16 groups x 16 rows) in A-16x4 layout ------
    // lane: M = r15; VGPR j holds K = 2*kh + j; chunk kc covers rows 4*kc..+3
    v2f A[4];
#pragma unroll
    for (int kc = 0; kc < 4; ++kc) {
#pragma unroll
        for (int j = 0; j < 2; ++j) {
            const int row   = 4 * kc + 2 * kh + j;
            const int rrank = __shfl(rank, row, 32);
            A[kc][j] = (rrank == r15) ? inv : 0.0f;
        }
    }

    // group ids for the 8 D-matrix rows this lane will hold (C/D layout:
    // VGPR v -> M = v + 8*half, N = lane%16)
    int  gid[8];
    bool gok[8];
#pragma unroll
    for (int v = 0; v < 8; ++v) {
        const int m = v + 8 * kh;
        gid[v] = __shfl(myGroup, m, 32);
        gok[v] = (m < nm);
    }

    const float* fbase = feats + (size_t)(b * Sdim + rowBase) * Hdim;
    float*       obase = out   + (size_t)b * Gdim * Hdim;

    // --- stream H in 16-column tiles; D = P x F via 4 chained K=4 WMMAs -----
    for (int t = 0; t < Hdim / 16; ++t) {
        const int hcol = t * 16 + r15;            // N = lane%16
        v8f c = {};
#pragma unroll
        for (int kc = 0; kc < 4; ++kc) {
            // B 4x16 layout (mirrors A/C convention): lane: N = r15,
            // VGPR j holds K = 2*kh + j -> window row 4*kc + 2*kh + j
            const int r0 = 4 * kc + 2 * kh;
            v2f Bv;
            // feats are single-use: non-temporal keeps out lines hot in L2
            Bv[0] = __builtin_nontemporal_load(&fbase[(size_t)(r0 + 0) * Hdim + hcol]);
            Bv[1] = __builtin_nontemporal_load(&fbase[(size_t)(r0 + 1) * Hdim + hcol]);
            c = __builtin_amdgcn_wmma_f32_16x16x4_f32(
                    /*neg_a=*/false, A[kc], /*neg_b=*/false, Bv,
                    /*c_mod=*/(short)0, c, /*reuse_a=*/false, /*reuse_b=*/false);
        }
#pragma unroll
        for (int v = 0; v < 8; ++v) {
            if (gok[v]) {
                atomicAdd(&obase[(size_t)gid[v] * Hdim + hcol], c[v]);
            }
        }
    }
}

extern "C" void kernel_launch(void* const* d_in, const int* in_sizes, int n_in,
                              void* d_out, int out_size, void* d_ws, size_t ws_size,
                              hipStream_t stream) {
    const float* feats = (const float*)d_in[0];   // [B,S,H] f32
    const int*   seg   = (const int*)d_in[1];     // [B,S]   i32 (sorted per row)
    float*       out   = (float*)d_out;           // [B,G,H] f32
    float*       cnts  = (float*)d_ws;            // [B,G]   f32 scratch (32 KB)

    const int nOut = Bdim * Gdim * Hdim;          // == out_size (2,097,152)
    const int nCnt = Bdim * Gdim;

    grouping_zero_kernel<<<512, 256, 0, stream>>>(out, cnts, nOut, nCnt);

    grouping_counts_kernel<<<(Bdim * Sdim) / 256, 256, 0, stream>>>(seg, cnts);

    grouping_segmean_wmma_kernel<<<WINDOWS / WAVES_PER_BLOCK, 256, 0, stream>>>(
        feats, seg, out, cnts);
}